// GCN_72627896975870
// MI455X (gfx1250) — compile-verified
//
#include <hip/hip_runtime.h>

typedef float v2f __attribute__((ext_vector_type(2)));
typedef float v4f __attribute__((ext_vector_type(4)));
typedef float v8f __attribute__((ext_vector_type(8)));

#define N_NODES   100000
#define N_EDGES   1600000
#define D_FEAT    256
#define HIDDEN    256
#define N_CLASSES 64

// ---------------------------------------------------------------------------
// Degree / normalization
// ---------------------------------------------------------------------------
__global__ void k_deg_init(float* __restrict__ deg) {
    int i = blockIdx.x * blockDim.x + threadIdx.x;
    if (i < N_NODES) deg[i] = 1.0f;  // self loop counted up front
}

__global__ void k_deg_count(const int* __restrict__ col, float* __restrict__ deg) {
    int e = blockIdx.x * blockDim.x + threadIdx.x;
    if (e < N_EDGES) atomicAdd(&deg[col[e]], 1.0f);
}

__global__ void k_dinv(float* __restrict__ deg) {
    int i = blockIdx.x * blockDim.x + threadIdx.x;
    if (i < N_NODES) deg[i] = rsqrtf(deg[i]);  // deg >= 1 always
}

// ---------------------------------------------------------------------------
// Pack W[K x NCOLS] into WMMA fragment order so each lane's B fragments for
// two adjacent N-tiles are 16 contiguous bytes (one global_load_b128 feeds
// two v_wmma_f32_16x16x4_f32).
//
// Fragment for (kb, t, lane): { W[kb*4 + 2*half][t*16 + m],
//                               W[kb*4 + 2*half + 1][t*16 + m] }
// Packed address (floats):  (((kb*(NT/2) + t/2)*32 + lane)*4) + (t&1)*2
// ---------------------------------------------------------------------------
template <int NCOLS, int K>
__global__ void k_pack_w(const float* __restrict__ W, float* __restrict__ Wp) {
    constexpr int NT = NCOLS / 16;
    int idx = blockIdx.x * blockDim.x + threadIdx.x;
    if (idx >= (K / 4) * NT * 32) return;
    int lane = idx & 31;
    int t    = (idx >> 5) % NT;
    int kb   = idx / (32 * NT);
    int m = lane & 15, half = lane >> 4;
    int ka = kb * 4 + 2 * half;
    int n  = t * 16 + m;
    size_t o = ((((size_t)kb * (NT / 2) + (t >> 1)) * 32 + lane) * 4) + (size_t)(t & 1) * 2;
    Wp[o]     = W[(size_t)ka * NCOLS + n];
    Wp[o + 1] = W[(size_t)(ka + 1) * NCOLS + n];
}

// ---------------------------------------------------------------------------
// fp32 WMMA GEMM: C[M x NCOLS] = A[M x K] * Wp(packed B)
// One wave owns a 16-row M-tile across all NCOLS/16 N-tiles; A fragment is
// loaded once per k-step (b64) and reused by NT WMMAs; B comes from the
// packed layout as b128 loads (2 fragments per load).
//
// ISA layouts (CDNA5 ISA):
//   A 16x4  : lane&15 = M row; (lane>>4) selects K pair; VGPR0/1 = K, K+1
//   B 4x16  : lane&15 = N col; (lane>>4) selects K pair; VGPR0/1 = K, K+1
//   C/D     : VGPR j holds M = j + 8*(lane>>4), N = lane&15
// ---------------------------------------------------------------------------
template <int NCOLS, int K>
__global__ __launch_bounds__(256)
void k_gemm_wmma(const float* __restrict__ A, const float* __restrict__ Wp,
                 float* __restrict__ C, int M) {
    constexpr int NT = NCOLS / 16;
    const int wave = (blockIdx.x * blockDim.x + threadIdx.x) >> 5;  // uniform per wave
    const int lane = threadIdx.x & 31;
    const int m0 = wave * 16;
    if (m0 >= M) return;  // whole-wave uniform exit; EXEC stays all-ones for WMMA

    const int m    = lane & 15;
    const int half = lane >> 4;

    v8f acc[NT] = {};

    const float* arow = A + (size_t)(m0 + m) * K + 2 * half;
    const v4f*   bp0  = (const v4f*)Wp + lane;

    #pragma unroll 2
    for (int kb = 0; kb < K / 4; ++kb) {
        const v2f a = *(const v2f*)(arow + kb * 4);       // global_load_b64
        const v4f* bp = bp0 + (size_t)kb * (NT / 2) * 32;
        #pragma unroll
        for (int tp = 0; tp < NT / 2; ++tp) {
            const v4f b4 = bp[(size_t)tp * 32];           // global_load_b128
            v2f b0, b1;
            b0.x = b4.x; b0.y = b4.y;
            b1.x = b4.z; b1.y = b4.w;
            acc[2 * tp] = __builtin_amdgcn_wmma_f32_16x16x4_f32(
                false, a, false, b0, (short)0, acc[2 * tp], false, false);
            acc[2 * tp + 1] = __builtin_amdgcn_wmma_f32_16x16x4_f32(
                false, a, false, b1, (short)0, acc[2 * tp + 1], false, false);
        }
    }

    #pragma unroll
    for (int t = 0; t < NT; ++t) {
        #pragma unroll
        for (int j = 0; j < 8; ++j) {
            C[(size_t)(m0 + j + 8 * half) * NCOLS + t * 16 + m] = acc[t][j];
        }
    }
}

// ---------------------------------------------------------------------------
// Self-loop + bias init:  agg[i][f] = h[i][f] * dinv[i]^2 + bias[f]
// ---------------------------------------------------------------------------
template <int F>
__global__ void k_selfloop_bias(const float* __restrict__ h,
                                const float* __restrict__ dinv,
                                const float* __restrict__ bias,
                                float* __restrict__ agg) {
    size_t i = (size_t)blockIdx.x * blockDim.x + threadIdx.x;
    if (i < (size_t)N_NODES * F) {
        int node = (int)(i / F);
        int f    = (int)(i % F);
        float d = dinv[node];
        agg[i] = h[i] * d * d + bias[f];
    }
}

// ---------------------------------------------------------------------------
// Edge scatter: agg[col] += h[row] * dinv[row]*dinv[col]; one wave per edge.
// agg fits in the 192MB L2, so these atomics stay on-chip. The strided
// (f + lane) pattern keeps each atomic instruction covering one 128B line.
// ---------------------------------------------------------------------------
template <int F>
__global__ __launch_bounds__(256)
void k_scatter(const int* __restrict__ row, const int* __restrict__ col,
               const float* __restrict__ dinv, const float* __restrict__ h,
               float* __restrict__ agg) {
    const int gw   = (blockIdx.x * blockDim.x + threadIdx.x) >> 5;
    const int lane = threadIdx.x & 31;
    const int nw   = (gridDim.x * blockDim.x) >> 5;
    for (int e = gw; e < N_EDGES; e += nw) {
        const int r = row[e];
        const int c = col[e];
        const float norm = dinv[r] * dinv[c];
        const float* hr = h   + (size_t)r * F;
        float*       ac = agg + (size_t)c * F;
        #pragma unroll
        for (int f = 0; f < F; f += 32) {
            atomicAdd(&ac[f + lane], hr[f + lane] * norm);
        }
    }
}

// ---------------------------------------------------------------------------
// In-place ReLU
// ---------------------------------------------------------------------------
__global__ void k_relu(float* __restrict__ x, size_t n) {
    size_t i = (size_t)blockIdx.x * blockDim.x + threadIdx.x;
    if (i < n) x[i] = fmaxf(x[i], 0.0f);
}

// ---------------------------------------------------------------------------
// Launch
// ---------------------------------------------------------------------------
extern "C" void kernel_launch(void* const* d_in, const int* in_sizes, int n_in,
                              void* d_out, int out_size, void* d_ws, size_t ws_size,
                              hipStream_t stream) {
    const float* x    = (const float*)d_in[0];
    const int*   eidx = (const int*)d_in[1];   // [2, N_EDGES] flat
    const float* W1   = (const float*)d_in[2];
    const float* b1   = (const float*)d_in[3];
    const float* W2   = (const float*)d_in[4];
    const float* b2   = (const float*)d_in[5];
    float* out = (float*)d_out;

    const int* row = eidx;            // source nodes
    const int* col = eidx + N_EDGES;  // target nodes

    // Workspace carve-up (h2 reuses the h1 buffer once layer 1 is consumed)
    float* dinv = (float*)d_ws;                               // N_NODES (padded)
    float* h1   = dinv + ((N_NODES + 63) & ~63);              // N_NODES * 256
    float* agg1 = h1 + (size_t)N_NODES * HIDDEN;              // N_NODES * 256
    float* Wp1  = agg1 + (size_t)N_NODES * HIDDEN;            // 256*256 packed
    float* Wp2  = Wp1 + (size_t)D_FEAT * HIDDEN;              // 256*64 packed
    float* h2   = h1;                                         // N_NODES * 64 (reuse)

    const int T = 256;

    // 1) Symmetric normalization coefficients
    k_deg_init<<<(N_NODES + T - 1) / T, T, 0, stream>>>(dinv);
    k_deg_count<<<(N_EDGES + T - 1) / T, T, 0, stream>>>(col, dinv);
    k_dinv<<<(N_NODES + T - 1) / T, T, 0, stream>>>(dinv);

    // 2) Pack weights into WMMA fragment order (one-shot, tiny)
    k_pack_w<HIDDEN, D_FEAT><<<(D_FEAT / 4 * (HIDDEN / 16) * 32 + T - 1) / T, T, 0, stream>>>(W1, Wp1);
    k_pack_w<N_CLASSES, HIDDEN><<<(HIDDEN / 4 * (N_CLASSES / 16) * 32 + T - 1) / T, T, 0, stream>>>(W2, Wp2);

    // 3) h1 = x @ W1  (fp32 WMMA; 6250 M-tiles, 8 waves/block)
    const int mtiles = N_NODES / 16;                 // 6250
    const int gblocks = (mtiles + 7) / 8;            // 782
    k_gemm_wmma<HIDDEN, D_FEAT><<<gblocks, T, 0, stream>>>(x, Wp1, h1, N_NODES);

    // 4) agg1 = h1*dinv^2 + b1 ; scatter edges ; relu
    k_selfloop_bias<HIDDEN><<<(int)(((size_t)N_NODES * HIDDEN + T - 1) / T), T, 0, stream>>>(
        h1, dinv, b1, agg1);
    k_scatter<HIDDEN><<<2048, T, 0, stream>>>(row, col, dinv, h1, agg1);
    k_relu<<<(int)(((size_t)N_NODES * HIDDEN + T - 1) / T), T, 0, stream>>>(
        agg1, (size_t)N_NODES * HIDDEN);

    // 5) h2 = relu(agg1) @ W2
    k_gemm_wmma<N_CLASSES, HIDDEN><<<gblocks, T, 0, stream>>>(agg1, Wp2, h2, N_NODES);

    // 6) out = h2*dinv^2 + b2 ; scatter edges (direct into d_out)
    k_selfloop_bias<N_CLASSES><<<(int)(((size_t)N_NODES * N_CLASSES + T - 1) / T), T, 0, stream>>>(
        h2, dinv, b2, out);
    k_scatter<N_CLASSES><<<2048, T, 0, stream>>>(row, col, dinv, h2, out);
}